// WaveletStem_75127567942000
// MI455X (gfx1250) — compile-verified
//
#include <hip/hip_runtime.h>
#include <hip/hip_bf16.h>

typedef __attribute__((ext_vector_type(2))) float v2f;
typedef __attribute__((ext_vector_type(8))) float v8f;

#define TPB 256

// ---- LDS layout (floats) ----
// Union region U: phase1 y[12][6][66] (4752 f)  /  phase3 star[32][260] (8320 f)
#define S_U       0
#define STAR_STRIDE 260
#define U_SIZE    (32 * STAR_STRIDE)          // 8320
// sbuf: B-matrix staging, 8 rows (K padded) x 256 spatial
#define S_SB      (S_U + U_SIZE)              // 8320
#define SB_SIZE   (8 * 256)                   // 2048
// weights
#define S_W       (S_SB + SB_SIZE)            // 10368
#define W_DW1     (S_W + 0)    // 108
#define W_DW2     (S_W + 108)  // 108
#define W_PW1     (S_W + 216)  // 72
#define W_PW2     (S_W + 288)  // 72
#define W_BN1S    (S_W + 360)  // 6
#define W_BN1B    (S_W + 366)  // 6
#define W_BN2S    (S_W + 372)  // 6
#define W_BN2B    (S_W + 378)  // 6
#define W_FUSP    (S_W + 384)  // 32x8 padded = 256
#define W_FUSS    (S_W + 640)  // 32
#define W_FUSB    (S_W + 672)  // 32
#define W_THRHH   (S_W + 704)  // 3
#define W_SIGLL   (S_W + 707)
#define W_THRLH   (S_W + 708)
#define SMEM_F    (S_W + 712)                 // 11080 floats = 44320 B

// Level-2 tile geometry: 64 wide x 4 tall (=256 points) -> final tile 32x2.
#define TW2 64
#define TH2 4
#define YW  (TW2 + 2)          // 66
#define YH  (TH2 + 2)          // 6
#define YCH (YH * YW)          // 396 per (channel,subband)

__device__ __forceinline__ float relu6f(float v) {
    return fminf(fmaxf(v, 0.f), 6.f);
}
__device__ __forceinline__ float softthresh(float v, float t) {
    float m = fabsf(v) - t;
    return (m > 0.f) ? copysignf(m, v) : 0.f;
}
__device__ __forceinline__ float softplusf(float v) {
    return (v > 20.f) ? v : log1pf(expf(v));
}
// Haar 2x2 filter fi applied to [a b; c d] (all coeffs +-0.25)
__device__ __forceinline__ float haar4(int fi, float a, float b, float c, float d) {
    float sb = (fi >= 2) ? -b : b;                 // fi 2,3: -b
    float sc = (fi == 1 || fi == 3) ? -c : c;      // fi 1,3: -c
    float sd = (fi == 1 || fi == 2) ? -d : d;      // fi 1,2: -d
    return 0.25f * (a + sb + sc + sd);
}

__global__ __launch_bounds__(TPB)
void wavelet_stem_fused(const float* __restrict__ x,
                        const float* __restrict__ ll_alpha,
                        const float* __restrict__ lh_hl_thr,
                        const float* __restrict__ hh_thr,
                        const float* __restrict__ dw1_w,
                        const float* __restrict__ pw1_w,
                        const float* __restrict__ bn1_g,
                        const float* __restrict__ bn1_b,
                        const float* __restrict__ dw2_w,
                        const float* __restrict__ pw2_w,
                        const float* __restrict__ bn2_g,
                        const float* __restrict__ bn2_b,
                        const float* __restrict__ fus_w,
                        const float* __restrict__ fus_g,
                        const float* __restrict__ fus_b,
                        float* __restrict__ out) {
    __shared__ float sm[SMEM_F];
    const int tid = threadIdx.x;

    // Block -> (image, level-2 tile). Tiles: 4 across (64 wide), 64 down (4 tall).
    const int tile = blockIdx.x & 255;
    const int img  = blockIdx.x >> 8;
    const int tH0 = (tile >> 2) * TH2;  // level-2 row origin (0..252 step 4)
    const int tW0 = (tile & 3) * TW2;   // level-2 col origin (0,64,128,192)

    const float RSQ = rsqrtf(1.f + 1e-5f);

    // ---------------- Phase 0: stage weights / derived scalars ----------------
    for (int i = tid; i < 108; i += TPB) { sm[W_DW1 + i] = dw1_w[i]; sm[W_DW2 + i] = dw2_w[i]; }
    for (int i = tid; i < 72;  i += TPB) { sm[W_PW1 + i] = pw1_w[i]; sm[W_PW2 + i] = pw2_w[i]; }
    if (tid < 6) {
        sm[W_BN1S + tid] = bn1_g[tid] * RSQ; sm[W_BN1B + tid] = bn1_b[tid];
        sm[W_BN2S + tid] = bn2_g[tid] * RSQ; sm[W_BN2B + tid] = bn2_b[tid];
    }
    {   // fus_w (32x6) zero-padded to (32x8) for two K=4 WMMA steps
        int o = tid >> 3, k = tid & 7;  // tid < 256 covers all
        sm[W_FUSP + tid] = (k < 6) ? fus_w[o * 6 + k] : 0.f;
    }
    if (tid < 32) { sm[W_FUSS + tid] = fus_g[tid] * RSQ; sm[W_FUSB + tid] = fus_b[tid]; }
    if (tid < 3)  sm[W_THRHH + tid] = softplusf(hh_thr[tid]);
    if (tid == 0) {
        sm[W_SIGLL] = 1.f / (1.f + expf(-ll_alpha[0]));
        sm[W_THRLH] = softplusf(lh_hl_thr[0]);
    }
    __syncthreads();

    // ---------------- Phase 1: denoised Haar-1 tile with halo -> y[12][6][66] ----------------
    // torch-compatible interleave: channel j=4c+s uses filter base[j/3] on input channel c=j/4.
    {
        const float sig = sm[W_SIGLL];
        const float tlh = sm[W_THRLH];
        for (int idx = tid; idx < 3 * YCH; idx += TPB) {
            int c   = idx / YCH;
            int rem = idx - c * YCH;
            int hy  = rem / YW;
            int wy  = rem - hy * YW;
            int gh = tH0 + hy - 1, gw = tW0 + wy - 1;
            float a = 0.f, b = 0.f, cc = 0.f, d = 0.f;
            if ((unsigned)gh < 256u && (unsigned)gw < 256u) {
                const float* xp = x + (((size_t)img * 3 + c) * 512 + 2 * (size_t)gh) * 512 + 2 * gw;
                a = xp[0]; b = xp[1]; cc = xp[512]; d = xp[513];
            }
            const float thh = sm[W_THRHH + c];
            #pragma unroll
            for (int s = 0; s < 4; ++s) {
                int j  = 4 * c + s;
                int fi = j / 3;
                float v = haar4(fi, a, b, cc, d);
                if (s == 0)      v *= sig;
                else if (s < 3)  v = softthresh(v, tlh);
                else             v = softthresh(v, thh);
                sm[S_U + j * YCH + hy * YW + wy] = v;
            }
        }
    }
    __syncthreads();

    // ---------------- Phase 2: dw3x3 + pw + BN + relu6, both branches, product -> sbuf[8][256] ----------------
    {
        const int p  = tid;
        const int h2 = p >> 6, w2 = p & 63;      // interior point; halo offset (+1,+1) cancels
        float hid1[6] = {0,0,0,0,0,0};
        float hid2[6] = {0,0,0,0,0,0};
        #pragma unroll
        for (int j = 0; j < 12; ++j) {
            const float* yj = &sm[S_U + j * YCH + h2 * YW + w2];
            const float* f1 = &sm[W_DW1 + j * 9];
            const float* f2 = &sm[W_DW2 + j * 9];
            float a1 = 0.f, a2 = 0.f;
            #pragma unroll
            for (int dy = 0; dy < 3; ++dy) {
                #pragma unroll
                for (int dx = 0; dx < 3; ++dx) {
                    float v = yj[dy * YW + dx];
                    a1 = fmaf(f1[dy * 3 + dx], v, a1);
                    a2 = fmaf(f2[dy * 3 + dx], v, a2);
                }
            }
            #pragma unroll
            for (int h = 0; h < 6; ++h) {
                hid1[h] = fmaf(sm[W_PW1 + h * 12 + j], a1, hid1[h]);
                hid2[h] = fmaf(sm[W_PW2 + h * 12 + j], a2, hid2[h]);
            }
        }
        #pragma unroll
        for (int h = 0; h < 6; ++h) {
            float u1 = relu6f(fmaf(sm[W_BN1S + h], hid1[h], sm[W_BN1B + h]));
            float u2 = relu6f(fmaf(sm[W_BN2S + h], hid2[h], sm[W_BN2B + h]));
            sm[S_SB + h * 256 + p] = u1 * u2;
        }
        sm[S_SB + 6 * 256 + p] = 0.f;   // K padding rows
        sm[S_SB + 7 * 256 + p] = 0.f;
    }
    __syncthreads();   // sbuf ready; y region free for reuse

    // ---------------- Phase 3: fuse 1x1 (6->32) via V_WMMA_F32_16X16X4_F32, BN + relu6 -> star ----------------
    {
        const int wave  = tid >> 5;
        const int lane  = tid & 31;
        const int laneN = lane & 15;
        const int hi    = (lane >= 16) ? 1 : 0;
        #pragma unroll
        for (int chunk = 0; chunk < 2; ++chunk) {
            const int pbase = wave * 32 + chunk * 16;
            #pragma unroll
            for (int m = 0; m < 2; ++m) {
                v8f acc = {0.f, 0.f, 0.f, 0.f, 0.f, 0.f, 0.f, 0.f};
                #pragma unroll
                for (int s = 0; s < 2; ++s) {
                    // 32-bit A 16x4 layout: lanes 0-15 hold K=k0,k0+1; lanes 16-31 hold K=k0+2,k0+3
                    const int k0 = 4 * s + 2 * hi;
                    v2f a, b;
                    a.x = sm[W_FUSP + (m * 16 + laneN) * 8 + k0];
                    a.y = sm[W_FUSP + (m * 16 + laneN) * 8 + k0 + 1];
                    b.x = sm[S_SB + k0 * 256 + pbase + laneN];
                    b.y = sm[S_SB + (k0 + 1) * 256 + pbase + laneN];
                    acc = __builtin_amdgcn_wmma_f32_16x16x4_f32(
                        false, a, false, b, (short)0, acc, false, false);
                }
                #pragma unroll
                for (int r = 0; r < 8; ++r) {
                    const int M = m * 16 + r + 8 * hi;   // C/D layout: vgpr r, lane-half selects M/M+8
                    float v = relu6f(fmaf(sm[W_FUSS + M], acc[r], sm[W_FUSB + M]));
                    sm[S_U + M * STAR_STRIDE + pbase + laneN] = v;
                }
            }
        }
    }
    __syncthreads();

    // ---------------- Phase 4: Haar-2 (filter o/32, channel o/4 interleave) -> global ----------------
    // Final tile is 32 wide x 2 tall: each (oc,h4) row = 32 consecutive floats = full 128B line.
    {
        const int H4b = tH0 >> 1, W4b = tW0 >> 1;
        for (int i = tid; i < 128 * 2 * 32; i += TPB) {
            const int oc  = i >> 6;          // 0..127
            const int h4  = (i >> 5) & 1;    // 0..1
            const int w4  = i & 31;          // 0..31
            const int mid = oc >> 2;         // input channel of Haar-2
            const int fi  = oc >> 5;         // filter index
            const float* sp = &sm[S_U + mid * STAR_STRIDE + (2 * h4) * TW2 + 2 * w4];
            float v = haar4(fi, sp[0], sp[1], sp[TW2], sp[TW2 + 1]);
            out[(((size_t)img * 128 + oc) * 128 + H4b + h4) * 128 + W4b + w4] = v;
        }
    }
}

extern "C" void kernel_launch(void* const* d_in, const int* in_sizes, int n_in,
                              void* d_out, int out_size, void* d_ws, size_t ws_size,
                              hipStream_t stream) {
    (void)in_sizes; (void)n_in; (void)out_size; (void)d_ws; (void)ws_size;
    const float* x        = (const float*)d_in[0];
    const float* ll_alpha = (const float*)d_in[1];
    const float* lh_hl    = (const float*)d_in[2];
    const float* hh_thr   = (const float*)d_in[3];
    const float* dw1_w    = (const float*)d_in[4];
    const float* pw1_w    = (const float*)d_in[5];
    const float* bn1_g    = (const float*)d_in[6];
    const float* bn1_b    = (const float*)d_in[7];
    const float* dw2_w    = (const float*)d_in[8];
    const float* pw2_w    = (const float*)d_in[9];
    const float* bn2_g    = (const float*)d_in[10];
    const float* bn2_b    = (const float*)d_in[11];
    const float* fus_w    = (const float*)d_in[12];
    const float* fus_g    = (const float*)d_in[13];
    const float* fus_b    = (const float*)d_in[14];
    float* out = (float*)d_out;

    // 32 images x 256 tiles (64x4 level-2 tiles over 256x256)
    dim3 grid(32 * 256), block(TPB);
    hipLaunchKernelGGL(wavelet_stem_fused, grid, block, 0, stream,
                       x, ll_alpha, lh_hl, hh_thr,
                       dw1_w, pw1_w, bn1_g, bn1_b,
                       dw2_w, pw2_w, bn2_g, bn2_b,
                       fus_w, fus_g, fus_b, out);
}